// JointsMSELoss_16741782520462
// MI455X (gfx1250) — compile-verified
//
#include <hip/hip_runtime.h>
#include <hip/hip_bf16.h>
#include <float.h>

// Problem constants (from the reference): B=256, J=17, H=96, W=72
#define B_N    256
#define J_N    17
#define HW_N   6912          // 96*72
#define CHUNKS 1728          // HW_N / 4 (float4 chunks per slice)
#define NSLICE 4352          // B_N * J_N
#define BLK    256

typedef __attribute__((ext_vector_type(2))) float v2f;
typedef __attribute__((ext_vector_type(8))) float v8f;

// Full 32-lane FP32 sum using V_WMMA_F32_16X16X4_F32.
// A (16x4): VGPR0 = lane value -> A[m][0] (lanes 0-15), A[m][2] (lanes 16-31);
//           VGPR1 = 0         -> A[m][1] = A[m][3] = 0.
// B (4x16): all ones, so D[m][n] = sum_k A[m][k] = s[m] + s[m+16].
// D layout: VGPR r holds row M=r (lanes 0-15) / M=r+8 (lanes 16-31), so
// summing a lane's 8 D VGPRs gives rows 0..7 (lo half) / 8..15 (hi half);
// one xor-16 shuffle completes the 32-lane total (broadcast to all lanes).
__device__ __forceinline__ float wave_reduce_sum_f32(float v) {
    v2f a; a[0] = v;    a[1] = 0.0f;
    v2f b; b[0] = 1.0f; b[1] = 1.0f;
    v8f c = {};
    c = __builtin_amdgcn_wmma_f32_16x16x4_f32(
            /*neg_a=*/false, a, /*neg_b=*/false, b,
            /*c_mod=*/(short)0, c, /*reuse_a=*/false, /*reuse_b=*/false);
    float s = ((c[0] + c[1]) + (c[2] + c[3])) + ((c[4] + c[5]) + (c[6] + c[7]));
    s += __shfl_xor(s, 16, 32);
    return s;
}

// Kernel 1: one block per (b, j) slice. Streams 2 x 27 KB with b128 loads,
// accumulates sum of squared diffs in FP32, scales by w^2, writes one partial.
__global__ void __launch_bounds__(BLK)
slice_sq_sum_kernel(const float* __restrict__ outp,
                    const float* __restrict__ tgtp,
                    const float* __restrict__ wp,
                    float* __restrict__ partial) {
    const int slice = blockIdx.x;
    const int tid   = threadIdx.x;

    const float4* o4 = (const float4*)(outp + (size_t)slice * HW_N);
    const float4* t4 = (const float4*)(tgtp + (size_t)slice * HW_N);

    float acc = 0.0f;
    for (int c = tid; c < CHUNKS; c += BLK) {
        float4 o = o4[c];
        float4 t = t4[c];
        float d0 = o.x - t.x;
        float d1 = o.y - t.y;
        float d2 = o.z - t.z;
        float d3 = o.w - t.w;
        acc = fmaf(d0, d0, acc);
        acc = fmaf(d1, d1, acc);
        acc = fmaf(d2, d2, acc);
        acc = fmaf(d3, d3, acc);
    }

    // Intra-wave reduction via WMMA (EXEC is all ones: full block, no branch).
    float wsum = wave_reduce_sum_f32(acc);

    __shared__ float lds[BLK / 32];
    if ((tid & 31) == 0) lds[tid >> 5] = wsum;
    __syncthreads();

    if (tid == 0) {
        float s = 0.0f;
        #pragma unroll
        for (int i = 0; i < BLK / 32; ++i) s += lds[i];
        const float w = wp[slice];
        partial[slice] = s * w * w;   // fold per-(b,j) weight^2 here
    }
}

// Kernel 2: one block. losses[j] = sum_b partial[b*J+j] / (B*HW);
// then thread 0 selects top-k and writes mean of top-k.
__global__ void __launch_bounds__(BLK)
finalize_kernel(const float* __restrict__ partial,
                const int*   __restrict__ topk_ptr,
                float*       __restrict__ outp) {
    const int tid = threadIdx.x;          // tid == batch index b
    __shared__ float losses[J_N];
    __shared__ float lds[BLK / 32];
    const float invN = 1.0f / ((float)B_N * (float)HW_N);

    for (int j = 0; j < J_N; ++j) {
        const float v    = partial[tid * J_N + j];
        const float wsum = wave_reduce_sum_f32(v);   // all 256 threads active
        if ((tid & 31) == 0) lds[tid >> 5] = wsum;
        __syncthreads();
        if (tid == 0) {
            float s = 0.0f;
            #pragma unroll
            for (int i = 0; i < BLK / 32; ++i) s += lds[i];
            losses[j] = s * invN;
        }
        __syncthreads();
    }

    if (tid == 0) {
        int k = topk_ptr ? topk_ptr[0] : 8;
        if (k < 1)    k = 1;
        if (k > J_N)  k = J_N;
        float arr[J_N];
        #pragma unroll
        for (int j = 0; j < J_N; ++j) arr[j] = losses[j];
        float sum = 0.0f;
        for (int t = 0; t < k; ++t) {      // selection of top-k (17 values)
            int   mi = 0;
            float mv = arr[0];
            for (int j = 1; j < J_N; ++j)
                if (arr[j] > mv) { mv = arr[j]; mi = j; }
            sum += mv;
            arr[mi] = -FLT_MAX;
        }
        outp[0] = sum / (float)k;
    }
}

extern "C" void kernel_launch(void* const* d_in, const int* in_sizes, int n_in,
                              void* d_out, int out_size, void* d_ws, size_t ws_size,
                              hipStream_t stream) {
    (void)in_sizes; (void)n_in; (void)out_size; (void)ws_size;

    const float* outp   = (const float*)d_in[0];   // output  [B,J,H,W] f32
    const float* tgtp   = (const float*)d_in[1];   // target  [B,J,H,W] f32
    const float* wp     = (const float*)d_in[2];   // weights [B,J,1]   f32
    const int*   topk_p = (const int*)  d_in[3];   // top_k scalar (int)

    float* partial = (float*)d_ws;                 // NSLICE floats (17 KB)

    slice_sq_sum_kernel<<<NSLICE, BLK, 0, stream>>>(outp, tgtp, wp, partial);
    finalize_kernel<<<1, BLK, 0, stream>>>(partial, topk_p, (float*)d_out);
}